// DifferentiableKnnGraphLayer_70875550319403
// MI455X (gfx1250) — compile-verified
//
#include <hip/hip_runtime.h>
#include <stdint.h>

#define NN   2048
#define BB   8
#define KSEL 20
#define TAUI 2.0f          /* 1/TAU */
#define CLMP 5.0f
#define EPSV 1e-20f
#define TPB  256
#define EPT  8             /* NN / TPB */
#define NWAVE (TPB / 32)

typedef unsigned int u32x4 __attribute__((ext_vector_type(4)));
typedef int          i32x4 __attribute__((ext_vector_type(4)));
typedef int          i32x8 __attribute__((ext_vector_type(8)));

typedef __attribute__((address_space(3))) unsigned char lds_byte_t;

__device__ __forceinline__ unsigned lds_off(void* p) {
  // generic -> LDS addrspace cast; integer value is the byte offset inside
  // the workgroup's LDS allocation (what D#.lds_addr wants).
  return (unsigned)(unsigned long long)(lds_byte_t*)p;
}

// Issue one TDM DMA: contiguous row of `nelem` f32 from global -> LDS.
// D# layout per CDNA5 ISA ch.8 (group0: count/lds_addr/global_addr/type,
// group1: data_size=4B, tensor_dim0=tile_dim0=nelem, dim1=tile_dim1=1).
__device__ __forceinline__ void tdm_load_row_f32(const float* gsrc, void* ldst,
                                                 unsigned nelem) {
  unsigned long long ga = (unsigned long long)gsrc;
  u32x4 g0;
  g0.x = 1u;                                        // count=1, user mode
  g0.y = lds_off(ldst);                             // lds_addr
  g0.z = (unsigned)(ga & 0xFFFFFFFFu);              // global_addr[31:0]
  g0.w = (unsigned)((ga >> 32) & 0x1FFFFFFu)        // global_addr[56:32]
       | (2u << 30);                                // type=2 ("image")
  i32x8 g1;
  g1[0] = (int)(2u << 16);                          // wg_mask=0, data_size=4B
  g1[1] = (int)((nelem & 0xFFFFu) << 16);           // tensor_dim0[15:0]
  g1[2] = (int)(((nelem >> 16) & 0xFFFFu)           // tensor_dim0[31:16]
       | (1u << 16));                               // tensor_dim1 = 1
  g1[3] = (int)((nelem & 0xFFFFu) << 16);           // tile_dim0 = nelem
  g1[4] = 1;                                        // tile_dim1=1, tile_dim2=0
  g1[5] = (int)nelem;                               // tensor_dim0_stride
  g1[6] = 0;
  g1[7] = 0;
  i32x4 z4 = {0, 0, 0, 0};
#if __has_include(<hip/amd_detail/amd_gfx1250_TDM.h>)
  i32x8 z8 = {0, 0, 0, 0, 0, 0, 0, 0};              // clang-23 / therock form
  __builtin_amdgcn_tensor_load_to_lds(g0, g1, z4, z4, z8, 0);
#else
  __builtin_amdgcn_tensor_load_to_lds(g0, g1, z4, z4, 0);   // ROCm 7.2 form
#endif
}

// ---- wave32 + cross-wave reductions ----------------------------------------

__device__ __forceinline__ float wave_max_f(float v) {
#pragma unroll
  for (int m = 16; m; m >>= 1) v = fmaxf(v, __shfl_xor(v, m, 32));
  return v;
}
__device__ __forceinline__ float wave_sum_f(float v) {
#pragma unroll
  for (int m = 16; m; m >>= 1) v += __shfl_xor(v, m, 32);
  return v;
}
__device__ __forceinline__ unsigned long long wave_max_u64(unsigned long long v) {
#pragma unroll
  for (int m = 16; m; m >>= 1) {
    unsigned lo = __shfl_xor((unsigned)(v & 0xFFFFFFFFu), m, 32);
    unsigned hi = __shfl_xor((unsigned)(v >> 32), m, 32);
    unsigned long long o = ((unsigned long long)hi << 32) | lo;
    v = (o > v) ? o : v;
  }
  return v;
}

__device__ __forceinline__ float block_max_f(float v, float* red) {
  v = wave_max_f(v);
  int w = threadIdx.x >> 5;
  if ((threadIdx.x & 31) == 0) red[w] = v;
  __syncthreads();
  float r = red[0];
#pragma unroll
  for (int k = 1; k < NWAVE; ++k) r = fmaxf(r, red[k]);
  __syncthreads();
  return r;
}
__device__ __forceinline__ float block_sum_f(float v, float* red) {
  v = wave_sum_f(v);
  int w = threadIdx.x >> 5;
  if ((threadIdx.x & 31) == 0) red[w] = v;
  __syncthreads();
  float r = red[0];
#pragma unroll
  for (int k = 1; k < NWAVE; ++k) r += red[k];
  __syncthreads();
  return r;
}
__device__ __forceinline__ unsigned long long block_max_u64(unsigned long long v,
                                                            unsigned long long* red) {
  v = wave_max_u64(v);
  int w = threadIdx.x >> 5;
  if ((threadIdx.x & 31) == 0) red[w] = v;
  __syncthreads();
  unsigned long long r = red[0];
#pragma unroll
  for (int k = 1; k < NWAVE; ++k) r = (red[k] > r) ? red[k] : r;
  __syncthreads();
  return r;
}

// ---- main kernel: one block per (b, i) row ---------------------------------

__global__ __launch_bounds__(TPB)
void knn_rows_kernel(const float* __restrict__ logits,
                     const float* __restrict__ gumbel,
                     float* __restrict__ edge_w) {
  __shared__ float s_gum[NN];
  __shared__ float s_log[NN];
  __shared__ float s_redf[NWAVE];
  __shared__ unsigned long long s_red64[NWAVE];

  const int row = blockIdx.x;            // 0 .. B*N-1
  const int b   = row >> 11;             // row / N
  const int i   = row & (NN - 1);        // row % N
  const int t   = threadIdx.x;

  if (t < 32) {                          // wave 0 drives the TDM
    tdm_load_row_f32(gumbel + (((size_t)b * NN) + i) * NN, s_gum, NN);
    tdm_load_row_f32(logits + (size_t)i * NN, s_log, NN);
    __builtin_amdgcn_s_wait_tensorcnt(0);
  }
  __syncthreads();

  float g[EPT], onehot[EPT], khot[EPT];
#pragma unroll
  for (int j = 0; j < EPT; ++j) {
    float lg = s_log[j * TPB + t];
    float sc = CLMP * tanhf(lg * (1.0f / CLMP));   // soft_clip
    g[j]      = sc + s_gum[j * TPB + t];
    onehot[j] = 0.0f;
    khot[j]   = 0.0f;
  }

  // K iterations of masked softmax accumulation
  for (int it = 0; it < KSEL; ++it) {
    if (it) {
#pragma unroll
      for (int j = 0; j < EPT; ++j)
        g[j] += __logf(fmaxf(1.0f - onehot[j], EPSV));
    }
    float lm = -3.4e38f;
#pragma unroll
    for (int j = 0; j < EPT; ++j) lm = fmaxf(lm, g[j]);
    lm = block_max_f(lm, s_redf);

    float e[EPT];
    float ls = 0.0f;
#pragma unroll
    for (int j = 0; j < EPT; ++j) {
      e[j] = __expf((g[j] - lm) * TAUI);
      ls += e[j];
    }
    ls = block_sum_f(ls, s_redf);
    float inv = 1.0f / ls;
#pragma unroll
    for (int j = 0; j < EPT; ++j) {
      onehot[j] = e[j] * inv;
      khot[j]  += onehot[j];
    }
  }

  // top-K via 20 block arg-max rounds (khot >= 0 so float bits order == uint order)
  unsigned sel = 0;
  for (int r = 0; r < KSEL; ++r) {
    float bv = -1.0f;
    int   bj = 0;
#pragma unroll
    for (int j = 0; j < EPT; ++j)
      if (khot[j] > bv) { bv = khot[j]; bj = j; }
    unsigned long long pack =
        ((unsigned long long)__float_as_uint(bv) << 32)
        | (unsigned)(bj * TPB + t);
    pack = block_max_u64(pack, s_red64);
    unsigned gidx = (unsigned)(pack & 0xFFFFFFFFu);
    if ((gidx & (TPB - 1)) == (unsigned)t) {
      int j = (int)(gidx >> 8);          // gidx / TPB
      khot[j] = 0.0f;                    // exclude from later rounds
      sel |= (1u << j);
    }
  }

  // forward value of straight-through estimator == hard k-hot
  float* w = edge_w + (size_t)row * NN;
#pragma unroll
  for (int j = 0; j < EPT; ++j)
    w[j * TPB + t] = ((sel >> j) & 1u) ? 1.0f : 0.0f;
}

// edge_index = [tile(arange(N), N); repeat(arange(N), N)], written as float
__global__ void edge_index_kernel(float* __restrict__ out) {
  size_t e = (size_t)blockIdx.x * blockDim.x + threadIdx.x;
  const size_t total = (size_t)NN * NN;
  if (e < total) {
    out[e]         = (float)(e & (NN - 1));   // e % N
    out[total + e] = (float)(e >> 11);        // e / N
  }
}

extern "C" void kernel_launch(void* const* d_in, const int* in_sizes, int n_in,
                              void* d_out, int out_size, void* d_ws, size_t ws_size,
                              hipStream_t stream) {
  (void)in_sizes; (void)n_in; (void)out_size; (void)d_ws; (void)ws_size;
  const float* logits = (const float*)d_in[0];
  const float* gumbel = (const float*)d_in[1];
  // d_in[2] (x) and d_in[3] (emb) are unused by the reference output.
  float* out    = (float*)d_out;
  float* edge_w = out + (size_t)2 * NN * NN;

  const int ei_blocks = (int)(((size_t)NN * NN + TPB - 1) / TPB);
  edge_index_kernel<<<ei_blocks, TPB, 0, stream>>>(out);
  knn_rows_kernel<<<BB * NN, TPB, 0, stream>>>(logits, gumbel, edge_w);
}